// BioGNN_15272903704952
// MI455X (gfx1250) — compile-verified
//
#include <hip/hip_runtime.h>
#include <math.h>

typedef int   v4i __attribute__((ext_vector_type(4)));
typedef float v4f __attribute__((ext_vector_type(4)));

// Fast device transcendentals mapping straight to gfx1250 VALU trans ops:
//   v_log_f32 (log2), v_exp_f32 (2^x)
__device__ __forceinline__ float fast_log2(float x) {
  return __builtin_amdgcn_logf(x);
}
__device__ __forceinline__ float fast_exp2(float x) {
  return __builtin_amdgcn_exp2f(x);
}

// ---------------------------------------------------------------------------
// Zero the workspace accumulators (must run every call: harness doesn't
// re-poison, and we must be deterministic).
// ---------------------------------------------------------------------------
__global__ __launch_bounds__(256) void zero_ws_kernel(float* __restrict__ w,
                                                      int n_words) {
  int i = (blockIdx.x * blockDim.x + threadIdx.x) * 4;
  if (i + 4 <= n_words) {
    v4f z = {0.0f, 0.0f, 0.0f, 0.0f};
    *(v4f*)(w + i) = z;
  } else {
    for (; i < n_words; ++i) w[i] = 0.0f;
  }
}

// ---------------------------------------------------------------------------
// Edge scatter: c = k * x[src]^hill, atomically accumulated onto sum[dst];
// flags[dst] |= FLAG records incoming-edge structure (bit0=act, bit1=inh).
//
// Edge streams are read-once -> non-temporal (th:NT) loads keep the 192MB L2
// reserved for the hot x[] gather + accumulator atomics. Prefetch one
// cacheline per 8 lanes (8 lanes * 4 edges * 4B = 128B line) 64KB ahead
// via the gfx1250 global_prefetch path.
// ---------------------------------------------------------------------------
template <unsigned FLAG>
__global__ __launch_bounds__(256) void edge_scatter_kernel(
    const int* __restrict__ src, const int* __restrict__ dst,
    const float* __restrict__ kk, const float* __restrict__ hh,
    const float* __restrict__ x, float* __restrict__ sum,
    unsigned int* __restrict__ flags, int nE) {
  int t = blockIdx.x * blockDim.x + threadIdx.x;
  int e = t * 4;
  if (e + 4 <= nE) {
    const int PF = 16384;  // elements ahead (~64KB per stream)
    if (((threadIdx.x & 7) == 0) && (e + PF + 4 <= nE)) {
      __builtin_prefetch(src + e + PF, 0, 0);  // global_prefetch_b8
      __builtin_prefetch(dst + e + PF, 0, 0);
      __builtin_prefetch(kk + e + PF, 0, 0);
      __builtin_prefetch(hh + e + PF, 0, 0);
    }
    v4i s = __builtin_nontemporal_load((const v4i*)(src + e));
    v4i d = __builtin_nontemporal_load((const v4i*)(dst + e));
    v4f k = __builtin_nontemporal_load((const v4f*)(kk + e));
    v4f h = __builtin_nontemporal_load((const v4f*)(hh + e));
#pragma unroll
    for (int j = 0; j < 4; ++j) {
      float xv = x[s[j]];  // random gather, L2-resident (2MB array)
      float hj = h[j];
      // hill==2 fast path (uniform branch for this model); generic Hill via
      // exp2(h*log2(x)) is valid because x > 0 by construction.
      float p = (hj == 2.0f) ? xv * xv : fast_exp2(hj * fast_log2(xv));
      float c = k[j] * p;
      atomicAdd(&sum[d[j]], c);        // no-return global_atomic_add_f32 @ L2
      atomicOr(&flags[d[j]], FLAG);    // no-return global_atomic_or_b32 @ L2
    }
  } else {
    for (int j = e; j < nE; ++j) {
      float xv = x[src[j]];
      float hj = hh[j];
      float p = (hj == 2.0f) ? xv * xv : fast_exp2(hj * fast_log2(xv));
      atomicAdd(&sum[dst[j]], kk[j] * p);
      atomicOr(&flags[dst[j]], FLAG);
    }
  }
}

// ---------------------------------------------------------------------------
// Per-node finalize:
//   numerator = has_act ? act_sum : 1
//   dx        = has_any ? numerator / (1 + act_sum + inh_sum) : 0
//   out       = exp(log_nu)*dx - exp(log_decay)*x + exp(log_growth)
// exp(y) computed as 2^(y*log2(e)) via v_exp_f32.
// ---------------------------------------------------------------------------
__global__ __launch_bounds__(256) void finalize_kernel(
    const float* __restrict__ x, const float* __restrict__ act_sum,
    const float* __restrict__ inh_sum, const unsigned int* __restrict__ flags,
    const float* __restrict__ log_decay, const float* __restrict__ log_growth,
    const float* __restrict__ log_nu, float* __restrict__ out, int n) {
  const float LOG2E = 1.44269504088896340736f;
  int i = blockIdx.x * blockDim.x + threadIdx.x;
  if (i >= n) return;
  unsigned f = flags[i];
  float num = act_sum[i];
  float inh = inh_sum[i];
  float numerator = (f & 1u) ? num : 1.0f;
  float den = 1.0f + num + inh;
  float dx = (f != 0u) ? (numerator / den) : 0.0f;
  float r = fast_exp2(log_nu[i] * LOG2E) * dx -
            fast_exp2(log_decay[i] * LOG2E) * x[i] +
            fast_exp2(log_growth[i] * LOG2E);
  __builtin_nontemporal_store(r, out + i);  // write-once result, th:NT
}

// ---------------------------------------------------------------------------
// Launch. Inputs (setup_inputs order):
//  0:x 1:act_src 2:act_dst 3:act_k 4:act_hill
//  5:inh_src 6:inh_dst 7:inh_k 8:inh_hill
//  9:log_decay 10:log_growth 11:log_nu
// ---------------------------------------------------------------------------
extern "C" void kernel_launch(void* const* d_in, const int* in_sizes, int n_in,
                              void* d_out, int out_size, void* d_ws,
                              size_t ws_size, hipStream_t stream) {
  const float* x = (const float*)d_in[0];
  const int* act_src = (const int*)d_in[1];
  const int* act_dst = (const int*)d_in[2];
  const float* act_k = (const float*)d_in[3];
  const float* act_h = (const float*)d_in[4];
  const int* inh_src = (const int*)d_in[5];
  const int* inh_dst = (const int*)d_in[6];
  const float* inh_k = (const float*)d_in[7];
  const float* inh_h = (const float*)d_in[8];
  const float* log_decay = (const float*)d_in[9];
  const float* log_growth = (const float*)d_in[10];
  const float* log_nu = (const float*)d_in[11];
  float* out = (float*)d_out;

  const int N = in_sizes[0];
  const int nA = in_sizes[1];
  const int nI = in_sizes[5];

  // Workspace layout: [act_sum: N floats][inh_sum: N floats][flags: N u32]
  float* act_sum = (float*)d_ws;
  float* inh_sum = act_sum + N;
  unsigned int* flags = (unsigned int*)(inh_sum + N);

  const int n_words = 3 * N;
  const int zb = ((n_words + 3) / 4 + 255) / 256;
  zero_ws_kernel<<<zb, 256, 0, stream>>>((float*)d_ws, n_words);

  const int ab = ((nA + 3) / 4 + 255) / 256;
  edge_scatter_kernel<1u><<<ab, 256, 0, stream>>>(act_src, act_dst, act_k,
                                                  act_h, x, act_sum, flags, nA);
  const int ib = ((nI + 3) / 4 + 255) / 256;
  edge_scatter_kernel<2u><<<ib, 256, 0, stream>>>(inh_src, inh_dst, inh_k,
                                                  inh_h, x, inh_sum, flags, nI);

  const int fb = (N + 255) / 256;
  finalize_kernel<<<fb, 256, 0, stream>>>(x, act_sum, inh_sum, flags, log_decay,
                                          log_growth, log_nu, out, N);
}